// TransformerV5T_60516089201257
// MI455X (gfx1250) — compile-verified
//
#include <hip/hip_runtime.h>
#include <hip/hip_bf16.h>
#include <math.h>

// Problem constants (from reference): B=2, C=64, H=W=64
#define BB     2
#define CC     64
#define HH     64
#define WW     64
#define LL     4096        // H*W
#define KF     576         // C*9 unfold features
#define LDA    580         // LDS row stride (dwords), pad 4 -> conflict-free
#define LDS_SC 66          // score tile row stride

typedef __attribute__((ext_vector_type(2))) float v2f;
typedef __attribute__((ext_vector_type(8))) float v8f;
typedef __attribute__((ext_vector_type(4))) int   v4i;

// ---- CDNA5 async global->LDS copy (ASYNCcnt path), with portable fallback ----
#if defined(__gfx1250__) && __has_builtin(__builtin_amdgcn_global_load_async_to_lds_b128)
#define ASYNC_LDS 1
#endif

#if defined(ASYNC_LDS)
#if __has_builtin(__builtin_amdgcn_s_wait_asynccnt)
#define WAIT_ASYNC() __builtin_amdgcn_s_wait_asynccnt(0)
#else
#define WAIT_ASYNC() asm volatile("s_wait_asynccnt 0" ::: "memory")
#endif
#else
#define WAIT_ASYNC()
#endif

__device__ __forceinline__ void cp16_to_lds(const float* gsrc, float* ldst) {
#if defined(ASYNC_LDS)
  __builtin_amdgcn_global_load_async_to_lds_b128(
      (__attribute__((address_space(1))) v4i*)gsrc,
      (__attribute__((address_space(3))) v4i*)ldst, 0, 0);
#else
  *(float4*)ldst = *(const float4*)gsrc;
#endif
}

// branchless top-3 (strict > keeps earliest index on ties, matching lax.top_k)
__device__ __forceinline__ void top3_update(float s, int gi,
                                            float& v0, float& v1, float& v2,
                                            int& i0, int& i1, int& i2) {
  bool g0 = s > v0, g1 = s > v1, g2 = s > v2;
  float nv2 = g1 ? v1 : (g2 ? s : v2);
  int   ni2 = g1 ? i1 : (g2 ? gi : i2);
  float nv1 = g0 ? v0 : (g1 ? s : v1);
  int   ni1 = g0 ? i0 : (g1 ? gi : i1);
  v0 = g0 ? s : v0;  i0 = g0 ? gi : i0;
  v1 = nv1;  i1 = ni1;  v2 = nv2;  i2 = ni2;
}

// ---------------------------------------------------------------------------
// K1/K2: 3x3 conv (zero pad 1), optional ReLU.  One thread per output element.
// ---------------------------------------------------------------------------
__global__ void conv3x3_kernel(const float* __restrict__ in,
                               const float* __restrict__ w,
                               const float* __restrict__ bias,
                               float* __restrict__ out, int do_relu) {
  int idx = blockIdx.x * blockDim.x + threadIdx.x;
  if (idx >= BB * CC * LL) return;
  int p  = idx & (LL - 1);
  int co = (idx >> 12) & (CC - 1);
  int b  = idx >> 18;
  int y  = p >> 6;
  int x  = p & (WW - 1);

  float sum = bias[co];
  for (int ci = 0; ci < CC; ++ci) {
    const float* wr = w + ((co * CC + ci) * 9);
    const float* ir = in + ((size_t)(b * CC + ci) << 12);
    #pragma unroll
    for (int dy = 0; dy < 3; ++dy) {
      int yy = y + dy - 1;
      if ((unsigned)yy >= HH) continue;
      #pragma unroll
      for (int dx = 0; dx < 3; ++dx) {
        int xx = x + dx - 1;
        if ((unsigned)xx >= WW) continue;
        sum += ir[yy * WW + xx] * wr[dy * 3 + dx];
      }
    }
  }
  if (do_relu) sum = fmaxf(sum, 0.0f);
  out[idx] = sum;
}

// ---------------------------------------------------------------------------
// K3: build transposed unfold xuT[b][q][576] (reflect pad) + invn[b][q]
// ---------------------------------------------------------------------------
__global__ void unfold_kernel(const float* __restrict__ x,
                              float* __restrict__ xuT,
                              float* __restrict__ invn) {
  int bq = blockIdx.x * blockDim.x + threadIdx.x;
  if (bq >= BB * LL) return;
  int q = bq & (LL - 1);
  int b = bq >> 12;
  int y = q >> 6;
  int xx = q & (WW - 1);

  int ry[3], rx[3];
  #pragma unroll
  for (int i = 0; i < 3; ++i) {
    int t = y + i - 1;
    ry[i] = (t < 0) ? -t : ((t >= HH) ? (2 * HH - 2 - t) : t);
    t = xx + i - 1;
    rx[i] = (t < 0) ? -t : ((t >= WW) ? (2 * WW - 2 - t) : t);
  }

  float* dst = xuT + (size_t)bq * KF;
  float sq = 0.0f;
  for (int c = 0; c < CC; ++c) {
    const float* ir = x + ((size_t)(b * CC + c) << 12);
    #pragma unroll
    for (int i = 0; i < 3; ++i) {
      const float* row = ir + ry[i] * WW;
      #pragma unroll
      for (int j = 0; j < 3; ++j) {
        float v = row[rx[j]];
        dst[c * 9 + i * 3 + j] = v;
        sq += v * v;
      }
    }
  }
  invn[bq] = 1.0f / fmaxf(sqrtf(sq), 1e-12f);
}

// ---------------------------------------------------------------------------
// K4: fused Gram GEMM (f32 WMMA 16x16x4) + streaming top-3 per query row.
// Grid: (L/64, B).  256 threads = 8 waves.  Wave (wm, wn2) owns a 16x32
// sub-tile of the 64x64 score tile.  A block (64x576) + B tile (64x576)
// resident in LDS for the whole K loop.  B tiles staged with async-to-LDS.
// Top-3 scan parallelized 4 threads/row with branchless update + LDS merge.
// ---------------------------------------------------------------------------
__global__ void __launch_bounds__(256)
gram_topk_kernel(const float* __restrict__ xuT,
                 const float* __restrict__ invn,
                 float* __restrict__ Rmax2, int* __restrict__ Ridx2,
                 float* __restrict__ Rmax3, int* __restrict__ Ridx3) {
  extern __shared__ float smem[];
  float* Alds = smem;                       // 64*580
  float* Blds = smem + 64 * LDA;            // 64*580
  float* Sc   = smem + 2 * 64 * LDA;        // 64*66
  float* Inv  = Sc + 64 * LDS_SC;           // 64
  float* Cand = Inv + 64;                   // 256*6 (3 {val,idx} per thread)

  const int b   = blockIdx.y;
  const int m0  = blockIdx.x * 64;
  const int tid = threadIdx.x;
  const int lane = tid & 31;
  const int wv  = tid >> 5;
  const int wm  = wv & 3;    // row sub-tile (16 rows)
  const int wn2 = wv >> 2;   // 32-col half

  // ---- stage A block: 64 rows x 576 f32 (async direct-to-LDS) ----
  const float* Abase = xuT + ((size_t)(b * LL + m0)) * KF;
  for (int c = tid; c < 64 * (KF / 4); c += 256) {
    int m = c / (KF / 4), f4 = (c % (KF / 4)) * 4;
    cp16_to_lds(Abase + m * KF + f4, Alds + m * LDA + f4);
  }

  // running top-3 (thread tid<64 owns query row m0+tid)
  float v0 = -INFINITY, v1 = -INFINITY, v2 = -INFINITY;
  int   i0 = 0, i1 = 0, i2 = 0;

  // per-lane LDS fragment pointers
  const int koff  = 2 * (lane >> 4);
  const float* Ap  = Alds + (16 * wm + (lane & 15)) * LDA + koff;
  const float* Bp0 = Blds + (32 * wn2 + (lane & 15)) * LDA + koff;
  const float* Bp1 = Bp0 + 16 * LDA;

  for (int n0 = 0; n0 < LL; n0 += 64) {
    // ---- stage B tile: 64 key rows x 576 (async direct-to-LDS) ----
    const float* Bbase = xuT + ((size_t)(b * LL + n0)) * KF;
    for (int c = tid; c < 64 * (KF / 4); c += 256) {
      int n = c / (KF / 4), f4 = (c % (KF / 4)) * 4;
      cp16_to_lds(Bbase + n * KF + f4, Blds + n * LDA + f4);
    }
    if (tid < 64) Inv[tid] = invn[b * LL + n0 + tid];
    // prefetch next tile toward L2/WGP$ while we compute
    if (n0 + 64 < LL) {
      const char* nb = (const char*)(xuT + ((size_t)(b * LL + n0 + 64)) * KF);
      __builtin_prefetch(nb + tid * 576, 0, 1);
    }
    WAIT_ASYNC();
    __syncthreads();

    // ---- 16x32 accumulate over K=576 with f32 WMMA ----
    v8f acc0 = {};
    v8f acc1 = {};
    #pragma unroll 4
    for (int k = 0; k < KF; k += 4) {
      v2f a  = *(const v2f*)(Ap + k);
      v2f b0 = *(const v2f*)(Bp0 + k);
      v2f b1 = *(const v2f*)(Bp1 + k);
      acc0 = __builtin_amdgcn_wmma_f32_16x16x4_f32(false, a, false, b0,
                                                   (short)0, acc0, false, false);
      acc1 = __builtin_amdgcn_wmma_f32_16x16x4_f32(false, a, false, b1,
                                                   (short)0, acc1, false, false);
    }

    // ---- spill score tile to LDS (C layout: vgpr r -> row r + 8*hi) ----
    {
      int hi   = lane >> 4;
      int col  = 32 * wn2 + (lane & 15);
      int rowb = 16 * wm + 8 * hi;
      #pragma unroll
      for (int r = 0; r < 8; ++r) {
        Sc[(rowb + r) * LDS_SC + col]      = acc0[r];
        Sc[(rowb + r) * LDS_SC + col + 16] = acc1[r];
      }
    }
    __syncthreads();

    // ---- parallel partial top-3: 4 threads per row, 16 cols each ----
    {
      int row = tid >> 2, seg = tid & 3;
      float p0 = -INFINITY, p1 = -INFINITY, p2 = -INFINITY;
      int   j0 = 0, j1 = 0, j2 = 0;
      const float* srow = Sc + row * LDS_SC + seg * 16;
      const float* ivp  = Inv + seg * 16;
      #pragma unroll
      for (int n = 0; n < 16; ++n) {
        float s = srow[n] * ivp[n];
        top3_update(s, n0 + seg * 16 + n, p0, p1, p2, j0, j1, j2);
      }
      float* cp = Cand + tid * 6;
      cp[0] = p0; ((int*)cp)[1] = j0;
      cp[2] = p1; ((int*)cp)[3] = j1;
      cp[4] = p2; ((int*)cp)[5] = j2;
    }
    __syncthreads();

    // ---- merge 4 segment candidates/row in ascending-index order ----
    if (tid < 64) {
      #pragma unroll
      for (int seg = 0; seg < 4; ++seg) {
        const float* cp = Cand + (tid * 4 + seg) * 6;
        #pragma unroll
        for (int r = 0; r < 3; ++r) {
          float s = cp[2 * r];
          int  gi = ((const int*)cp)[2 * r + 1];
          top3_update(s, gi, v0, v1, v2, i0, i1, i2);
        }
      }
    }
    __syncthreads();
  }

  if (tid < 64) {
    int q = m0 + tid;
    float iq = invn[b * LL + q];
    Rmax2[b * LL + q] = v1 * iq;  Ridx2[b * LL + q] = i1;   // rank 1 (t=2)
    Rmax3[b * LL + q] = v2 * iq;  Ridx3[b * LL + q] = i2;   // rank 2 (t=3)
  }
}

// ---------------------------------------------------------------------------
// K5: gather selected patches, fold3, scale by R_max -> tex_in[b][2C][L]
// ---------------------------------------------------------------------------
__global__ void texture_gather_kernel(const float* __restrict__ xuT,
                                      const float* __restrict__ Rmax2,
                                      const int* __restrict__ Ridx2,
                                      const float* __restrict__ Rmax3,
                                      const int* __restrict__ Ridx3,
                                      float* __restrict__ tex_in) {
  int bq = blockIdx.x * blockDim.x + threadIdx.x;
  if (bq >= BB * LL) return;
  int q = bq & (LL - 1);
  int b = bq >> 12;
  int y = q >> 6;
  int x = q & (WW - 1);

  int nv = 0;
  int offv[9], k2v[9], k3v[9];
  #pragma unroll
  for (int i = 0; i < 3; ++i) {
    int yy = y + 1 - i;
    if ((unsigned)yy >= HH) continue;
    #pragma unroll
    for (int j = 0; j < 3; ++j) {
      int xj = x + 1 - j;
      if ((unsigned)xj >= WW) continue;
      int qp = b * LL + yy * WW + xj;
      offv[nv] = i * 3 + j;
      k2v[nv]  = Ridx2[qp];
      k3v[nv]  = Ridx3[qp];
      ++nv;
    }
  }
  float sc2 = Rmax2[bq] * (1.0f / 9.0f);
  float sc3 = Rmax3[bq] * (1.0f / 9.0f);

  for (int c = 0; c < CC; ++c) {
    float s2 = 0.0f, s3 = 0.0f;
    for (int u = 0; u < nv; ++u) {
      int fo = c * 9 + offv[u];
      s2 += xuT[((size_t)(b * LL + k2v[u])) * KF + fo];
      s3 += xuT[((size_t)(b * LL + k3v[u])) * KF + fo];
    }
    tex_in[((size_t)(b * 2 * CC + c) << 12) + q]      = s2 * sc2;
    tex_in[((size_t)(b * 2 * CC + CC + c) << 12) + q] = s3 * sc3;
  }
}

// ---------------------------------------------------------------------------
// K6: texture = 1x1 conv (2C -> C)
// ---------------------------------------------------------------------------
__global__ void conv1x1_tex_kernel(const float* __restrict__ tex_in,
                                   const float* __restrict__ wt1,
                                   const float* __restrict__ bt1,
                                   float* __restrict__ texture) {
  int idx = blockIdx.x * blockDim.x + threadIdx.x;
  if (idx >= BB * CC * LL) return;
  int p  = idx & (LL - 1);
  int co = (idx >> 12) & (CC - 1);
  int b  = idx >> 18;
  const float* wr = wt1 + co * (2 * CC);
  const float* ir = tex_in + ((size_t)(b * 2 * CC) << 12) + p;
  float sum = bt1[co];
  for (int ch = 0; ch < 2 * CC; ++ch) sum += wr[ch] * ir[(size_t)ch << 12];
  texture[idx] = sum;
}

// ---------------------------------------------------------------------------
// K7: y = 1x1 conv over concat(feature, x, texture) (3C -> C)
// ---------------------------------------------------------------------------
__global__ void conv1x1_out_kernel(const float* __restrict__ feature,
                                   const float* __restrict__ x,
                                   const float* __restrict__ texture,
                                   const float* __restrict__ wt2,
                                   const float* __restrict__ bt2,
                                   float* __restrict__ out) {
  int idx = blockIdx.x * blockDim.x + threadIdx.x;
  if (idx >= BB * CC * LL) return;
  int p  = idx & (LL - 1);
  int co = (idx >> 12) & (CC - 1);
  int b  = idx >> 18;
  const float* wr = wt2 + co * (3 * CC);
  size_t base = (size_t)(b * CC) << 12;
  float sum = bt2[co];
  for (int ch = 0; ch < CC; ++ch) sum += wr[ch]          * feature[base + ((size_t)ch << 12) + p];
  for (int ch = 0; ch < CC; ++ch) sum += wr[CC + ch]     * x[base + ((size_t)ch << 12) + p];
  for (int ch = 0; ch < CC; ++ch) sum += wr[2 * CC + ch] * texture[base + ((size_t)ch << 12) + p];
  out[idx] = sum;
}

// ---------------------------------------------------------------------------
extern "C" void kernel_launch(void* const* d_in, const int* in_sizes, int n_in,
                              void* d_out, int out_size, void* d_ws, size_t ws_size,
                              hipStream_t stream) {
  const float* x   = (const float*)d_in[0];
  const float* w1  = (const float*)d_in[1];
  const float* b1  = (const float*)d_in[2];
  const float* w2  = (const float*)d_in[3];
  const float* b2  = (const float*)d_in[4];
  const float* wt1 = (const float*)d_in[5];
  const float* bt1 = (const float*)d_in[6];
  const float* wt2 = (const float*)d_in[7];
  const float* bt2 = (const float*)d_in[8];
  float* out = (float*)d_out;

  // workspace carve-up (floats)
  float* ws      = (float*)d_ws;
  float* mid     = ws;                         // B*C*L
  float* feature = mid     + (size_t)BB * CC * LL;
  float* xuT     = feature + (size_t)BB * CC * LL;   // B*L*576
  float* invn    = xuT     + (size_t)BB * LL * KF;   // B*L
  float* Rmax2   = invn    + (size_t)BB * LL;
  float* Rmax3   = Rmax2   + (size_t)BB * LL;
  int*   Ridx2   = (int*)(Rmax3 + (size_t)BB * LL);
  int*   Ridx3   = Ridx2   + (size_t)BB * LL;
  float* tex_in  = (float*)(Ridx3 + (size_t)BB * LL); // B*2C*L
  float* texture = tex_in  + (size_t)BB * 2 * CC * LL;

  const int n_bcl = BB * CC * LL;

  conv3x3_kernel<<<(n_bcl + 255) / 256, 256, 0, stream>>>(x, w1, b1, mid, 1);
  conv3x3_kernel<<<(n_bcl + 255) / 256, 256, 0, stream>>>(mid, w2, b2, feature, 0);
  unfold_kernel<<<(BB * LL + 255) / 256, 256, 0, stream>>>(x, xuT, invn);

  // fused Gram + top-3: ~313 KB dynamic LDS per workgroup
  size_t smem = (size_t)(2 * 64 * LDA + 64 * LDS_SC + 64 + 256 * 6) * sizeof(float);
  (void)hipFuncSetAttribute((const void*)gram_topk_kernel,
                            hipFuncAttributeMaxDynamicSharedMemorySize, (int)smem);
  gram_topk_kernel<<<dim3(LL / 64, BB), 256, smem, stream>>>(
      xuT, invn, Rmax2, Ridx2, Rmax3, Ridx3);

  texture_gather_kernel<<<(BB * LL + 255) / 256, 256, 0, stream>>>(
      xuT, Rmax2, Ridx2, Rmax3, Ridx3, tex_in);
  conv1x1_tex_kernel<<<(n_bcl + 255) / 256, 256, 0, stream>>>(tex_in, wt1, bt1, texture);
  conv1x1_out_kernel<<<(n_bcl + 255) / 256, 256, 0, stream>>>(feature, x, texture,
                                                              wt2, bt2, out);
}